// Trans_83580063580165
// MI455X (gfx1250) — compile-verified
//
#include <hip/hip_runtime.h>
#include <math.h>

// ---------------- sizes ----------------
#define B_   8
#define L_   512
#define CIN  16
#define DM   512
#define NH   8
#define KD   128          // KDIM
#define HD_  16
#define FF_  16
#define TOPK_ 6
#define JI   12
#define KMAX_ 256
#define MROWS (L_ * B_)   // 4096

typedef __attribute__((ext_vector_type(2))) float v2f;
typedef __attribute__((ext_vector_type(8))) float v8f;

// ---------------- WMMA helper (fp32, 16x16x4) ----------------
__device__ __forceinline__ v8f wmma4(v2f a, v2f b, v8f c) {
#if __has_builtin(__builtin_amdgcn_wmma_f32_16x16x4_f32)
  return __builtin_amdgcn_wmma_f32_16x16x4_f32(false, a, false, b, (short)0, c,
                                               false, false);
#else
  // host pass / fallback only; never executed on gfx1250
  c[0] += a.x * b.x + a.y * b.y;
  return c;
#endif
}

// ---------- CDNA5 async global->LDS copy helpers (ASYNCcnt path) ----------
__device__ __forceinline__ void async_copy_b128(float* lds_dst, const float* gsrc) {
#if defined(__AMDGCN__)
  unsigned int loff = (unsigned int)(unsigned long long)lds_dst;  // LDS aperture low bits
  asm volatile("global_load_async_to_lds_b128 %0, %1, off"
               :: "v"(loff), "v"(gsrc) : "memory");
#else
  for (int j = 0; j < 4; j++) lds_dst[j] = gsrc[j];
#endif
}
__device__ __forceinline__ void wait_async_all() {
#if defined(__AMDGCN__)
  asm volatile("s_wait_asynccnt 0x0" ::: "memory");
#endif
}

// ---------------- generic GEMM (plain): C = A[M,K] @ Bw[N,K]^T + bias -----
__global__ void gemm_wmma(const float* __restrict__ A,
                          const float* __restrict__ Bw,
                          const float* __restrict__ bias,
                          float* __restrict__ C, int M, int N, int K) {
  int wid  = (blockIdx.x * blockDim.x + threadIdx.x) >> 5;
  int lane = threadIdx.x & 31;
  int ntiles = N >> 4;
  int total  = (M >> 4) * ntiles;
  if (wid >= total) return;
  int tm = wid / ntiles, tn = wid % ntiles;
  int half = lane >> 4, l15 = lane & 15;
  int arow = tm * 16 + l15;
  int bcol = tn * 16 + l15;
  const float* ap = A + (size_t)arow * K;
  const float* bp = Bw + (size_t)bcol * K;
  float bv = bias ? bias[bcol] : 0.f;
  v8f c;
#pragma unroll
  for (int v = 0; v < 8; v++) c[v] = bv;
  for (int k0 = 0; k0 < K; k0 += 4) {
    int ka = k0 + 2 * half;
    __builtin_prefetch(bp + ka + 64, 0, 1);
    v2f a, b;
    a.x = ap[ka]; a.y = ap[ka + 1];
    b.x = bp[ka]; b.y = bp[ka + 1];
    c = wmma4(a, b, c);
  }
  int crow0 = tm * 16 + 8 * half;
#pragma unroll
  for (int v = 0; v < 8; v++) C[(size_t)(crow0 + v) * N + bcol] = c[v];
}

// ------ GEMM with block-shared A strip staged in LDS via async copies -----
// Requires: ntiles (=N/16) multiple of 8, K <= 512, grid = tiles/8 blocks of 256.
// All 8 waves of a block share the same tm => same 16xK A strip (8x reuse).
__global__ void gemm_wmma_lds(const float* __restrict__ A,
                              const float* __restrict__ Bw,
                              const float* __restrict__ bias,
                              float* __restrict__ C, int M, int N, int K) {
  __shared__ float ldsA[16 * 512];
  int lane = threadIdx.x & 31;
  int ntiles = N >> 4;
  int gwid = blockIdx.x * 8 + (threadIdx.x >> 5);
  int tm = gwid / ntiles;          // uniform across the block (ntiles % 8 == 0)
  int tn = gwid % ntiles;
  // cooperative async stage of contiguous 16xK strip (16*K*4 bytes <= 32KB)
  const float* astrip = A + (size_t)(tm * 16) * K;
  int nchunks = (16 * K) >> 2;     // b128 chunks
  for (int ch = threadIdx.x; ch < nchunks; ch += blockDim.x)
    async_copy_b128(&ldsA[ch * 4], astrip + ch * 4);
  wait_async_all();                // this wave's asyncs landed in LDS
  __syncthreads();                 // all waves' asyncs visible

  int half = lane >> 4, l15 = lane & 15;
  int bcol = tn * 16 + l15;
  const float* la = &ldsA[l15 * K];            // A row for this lane (LDS)
  const float* bp = Bw + (size_t)bcol * K;     // B col (global)
  float bv = bias ? bias[bcol] : 0.f;
  v8f c;
#pragma unroll
  for (int v = 0; v < 8; v++) c[v] = bv;
  for (int k0 = 0; k0 < K; k0 += 4) {
    int ka = k0 + 2 * half;
    __builtin_prefetch(bp + ka + 64, 0, 1);
    v2f a, b;
    a.x = la[ka]; a.y = la[ka + 1];            // ds_load from staged strip
    b.x = bp[ka]; b.y = bp[ka + 1];
    c = wmma4(a, b, c);
  }
  int crow0 = tm * 16 + 8 * half;
#pragma unroll
  for (int v = 0; v < 8; v++) C[(size_t)(crow0 + v) * N + bcol] = c[v];
}

// ---------------- embed: conv1d(k=3, wrap) + pos-enc ----------------------
__global__ void embed_k(const float* __restrict__ src,
                        const float* __restrict__ tw,
                        const float* __restrict__ tb, float* __restrict__ x) {
  int idx = blockIdx.x * blockDim.x + threadIdx.x;
  if (idx >= MROWS * DM) return;
  int d = idx & 511;
  int m = idx >> 9;
  int b = m & 7;
  int l = m >> 3;
  float acc = tb[d];
#pragma unroll
  for (int kk = 0; kk < 3; kk++) {
    int ls = (l + kk + L_ - 1) & (L_ - 1);
    const float* sp = src + (b * L_ + ls) * CIN;
    const float* wp = tw + d * (CIN * 3) + kk;
#pragma unroll
    for (int c = 0; c < CIN; c++) acc += sp[c] * wp[c * 3];
  }
  int j2 = d & ~1;
  float freq = expf(-(float)j2 * (9.210340371976184f / (float)DM));
  float ang = (float)l * freq;
  acc += (d & 1) ? cosf(ang) : sinf(ang);
  x[idx] = acc;
}

// ---------------- repack (L,B,KD) -> (bh, hd, L) --------------------------
__global__ void repack_qk(const float* __restrict__ lin, float* __restrict__ T) {
  int idx = blockIdx.x * blockDim.x + threadIdx.x;
  if (idx >= 64 * HD_ * L_) return;
  int l = idx & 511;
  int i = (idx >> 9) & 15;
  int bh = idx >> 13;
  int b = bh >> 3, h = bh & 7;
  T[idx] = lin[(size_t)(l * 8 + b) * KD + h * HD_ + i];
}

// ------------- cos-sim of emb rows, top-12 + softmax ----------------------
__global__ void topcos_k(const float* __restrict__ emb, int* __restrict__ ti,
                         float* __restrict__ tw) {
  __shared__ float norms[HD_];
  int i = threadIdx.x;
  if (i < HD_) {
    float s = 0.f;
    for (int j = 0; j < 64; j++) { float v = emb[i * 64 + j]; s += v * v; }
    norms[i] = sqrtf(s);
  }
  __syncthreads();
  if (i < HD_) {
    float cosr[HD_];
    for (int j = 0; j < HD_; j++) {
      float d = 0.f;
      for (int c = 0; c < 64; c++) d += emb[i * 64 + c] * emb[j * 64 + c];
      cosr[j] = d / (norms[i] * norms[j]);
    }
    bool used[HD_];
    for (int j = 0; j < HD_; j++) used[j] = false;
    float tv[JI]; int tix[JI];
    for (int k = 0; k < JI; k++) {
      float best = -1e30f; int bi = 0;
      for (int j = 0; j < HD_; j++)
        if (!used[j] && cosr[j] > best) { best = cosr[j]; bi = j; }
      used[bi] = true; tv[k] = best; tix[k] = bi;
    }
    float mx = tv[0], se = 0.f, e[JI];
    for (int k = 0; k < JI; k++) { e[k] = expf(tv[k] - mx); se += e[k]; }
    for (int k = 0; k < JI; k++) {
      tw[i * JI + k] = e[k] / se;
      ti[i * JI + k] = tix[k];
    }
  }
}

// -------------- kw[bh,i,s] = sum_jj tw[i,jj] * kT[bh,ti[i,jj],s] ----------
__global__ void build_kw(const float* __restrict__ kT, const int* __restrict__ ti,
                         const float* __restrict__ tw, float* __restrict__ kw) {
  int idx = blockIdx.x * blockDim.x + threadIdx.x;
  if (idx >= 64 * HD_ * L_) return;
  int s = idx & 511;
  int i = (idx >> 9) & 15;
  int bh = idx >> 13;
  float a = 0.f;
#pragma unroll
  for (int jj = 0; jj < JI; jj++)
    a += tw[i * JI + jj] * kT[((size_t)bh * HD_ + ti[i * JI + jj]) * L_ + s];
  kw[idx] = a;
}

// -------- len_corr[bh,t] = (1/HD) sum_i sum_s q[(s+t)%L] * kw[s] ----------
__global__ void len_corr_k(const float* __restrict__ qT,
                           const float* __restrict__ kw,
                           float* __restrict__ lc) {
  int wid  = (blockIdx.x * blockDim.x + threadIdx.x) >> 5;
  int lane = threadIdx.x & 31;
  if (wid >= 64 * L_) return;
  int bh = wid >> 9, t = wid & 511;
  const float* qb = qT + (size_t)bh * HD_ * L_;
  const float* kb = kw + (size_t)bh * HD_ * L_;
  float acc = 0.f;
  for (int i = 0; i < HD_; i++) {
    const float* qi = qb + i * L_;
    const float* ki = kb + i * L_;
    for (int s = lane; s < L_; s += 32) acc += qi[(s + t) & (L_ - 1)] * ki[s];
  }
#pragma unroll
  for (int off = 16; off; off >>= 1) acc += __shfl_xor(acc, off);
  if (lane == 0) lc[wid] = acc * (1.f / (float)HD_);
}

// ---------- mean over bh, top-6 + softmax + folded sizes ------------------
__global__ void mean_topk(const float* __restrict__ lc, float* __restrict__ tmp6,
                          int* __restrict__ folded) {
  __shared__ float lm[L_];
  int t = threadIdx.x;
  float s = 0.f;
  for (int b = 0; b < 64; b++) s += lc[b * L_ + t];
  lm[t] = s * (1.f / 64.f);
  __syncthreads();
  if (t == 0) {
    bool used[L_];
    for (int j = 0; j < L_; j++) used[j] = false;
    float w[TOPK_]; int id[TOPK_];
    for (int k = 0; k < TOPK_; k++) {
      float best = -1e30f; int bi = 0;
      for (int j = 0; j < L_; j++)
        if (!used[j] && lm[j] > best) { best = lm[j]; bi = j; }
      used[bi] = true; w[k] = best; id[k] = bi;
    }
    float mx = w[0], se = 0.f, e[TOPK_];
    for (int k = 0; k < TOPK_; k++) { e[k] = expf(w[k] - mx); se += e[k]; }
    for (int k = 0; k < TOPK_; k++) {
      tmp6[k] = e[k] / se;
      folded[k] = (id[k] >= L_ / 2) ? (L_ - id[k]) : id[k];
    }
  }
}

// ---------- build vt (bh, 32, L) = concat(v, emb broadcast) transposed ----
__global__ void build_v(const float* __restrict__ v_lin,
                        const float* __restrict__ emb, float* __restrict__ vt) {
  int idx = blockIdx.x * blockDim.x + threadIdx.x;
  if (idx >= 64 * L_ * 32) return;
  int c = idx & 31;
  int l = (idx >> 5) & 511;
  int bh = idx >> 14;
  int b = bh >> 3, h = bh & 7;
  float val = (c < HD_) ? v_lin[(size_t)(l * 8 + b) * KD + h * HD_ + c]
                        : emb[(c - HD_) * 64 + bh];
  vt[((size_t)bh * 32 + c) * L_ + l] = val;
}

__global__ void zero_k(float* __restrict__ p, int n) {
  int i = blockIdx.x * blockDim.x + threadIdx.x;
  if (i < n) p[i] = 0.f;
}

// ---------------- TCN causal conv (implicit GEMM, WMMA) -------------------
// phase 0: out=h (n,o,t) <- relu(conv1(in)+b);  phase 1: agg(n,t,o) += tmp*relu(relu(conv2+b)+vt)
__global__ void tcn_conv(const float* __restrict__ in,
                         const float* __restrict__ w_all,
                         const float* __restrict__ b_all,
                         const float* __restrict__ vt, float* __restrict__ out,
                         const int* __restrict__ folded,
                         const float* __restrict__ tmp6, int i, int phase) {
  int ind  = folded[i];
  int wid  = (blockIdx.x * blockDim.x + threadIdx.x) >> 5;
  int lane = threadIdx.x & 31;
  if (wid >= 64 * 32 * 2) return;
  int ot = wid & 1;
  int tt = (wid >> 1) & 31;
  int n  = wid >> 6;
  int half = lane >> 4, l15 = lane & 15;
  int t0 = tt * 16, o0 = ot * 16;
  int o = o0 + l15;
  if (ind <= 1) {
    if (phase == 1) {
      float sc = tmp6[i];
#pragma unroll
      for (int v = 0; v < 8; v++) {
        int t = t0 + v + 8 * half;
        out[((size_t)n * L_ + t) * 32 + o] += sc * vt[((size_t)n * 32 + o) * L_ + t];
      }
    }
    return;
  }
  int ta = t0 + l15;  // output time for A-row of this lane
  float bias = b_all[i * 32 + o];
  v8f c;
#pragma unroll
  for (int v = 0; v < 8; v++) c[v] = bias;
  const float* wbase = w_all + ((size_t)(i * 32 + o) * 32) * KMAX_;
  for (int ci = 0; ci < 32; ci++) {
    const float* arow = in + ((size_t)n * 32 + ci) * L_;
    const float* brow = wbase + ci * KMAX_;
    for (int kk0 = 0; kk0 < ind; kk0 += 4) {
      int ka = kk0 + 2 * half;
      int t1 = ta - (ind - 1) + ka;
      v2f a, b;
      a.x = (ka < ind && t1 >= 0) ? arow[t1] : 0.f;
      a.y = (ka + 1 < ind && t1 + 1 >= 0) ? arow[t1 + 1] : 0.f;
      b.x = (ka < ind) ? brow[ka] : 0.f;
      b.y = (ka + 1 < ind) ? brow[ka + 1] : 0.f;
      c = wmma4(a, b, c);
    }
  }
  if (phase == 0) {
#pragma unroll
    for (int v = 0; v < 8; v++) {
      int t = t0 + v + 8 * half;
      out[((size_t)n * 32 + o) * L_ + t] = fmaxf(c[v], 0.f);
    }
  } else {
    float sc = tmp6[i];
#pragma unroll
    for (int v = 0; v < 8; v++) {
      int t = t0 + v + 8 * half;
      float y  = fmaxf(c[v], 0.f);
      float y2 = fmaxf(y + vt[((size_t)n * 32 + o) * L_ + t], 0.f);
      out[((size_t)n * L_ + t) * 32 + o] += sc * y2;
    }
  }
}

// --------- agg (bh,l,32) -> aggR (m=l*8+b, h*32+c) ------------------------
__global__ void repack_agg(const float* __restrict__ agg, float* __restrict__ aggR) {
  int idx = blockIdx.x * blockDim.x + threadIdx.x;
  if (idx >= MROWS * 256) return;
  int k = idx & 255;
  int m = idx >> 8;
  int l = m >> 3, b = m & 7;
  int h = k >> 5, c = k & 31;
  aggR[idx] = agg[(((size_t)(b * 8 + h) * L_) + l) * 32 + c];
}

// ---------------- (optional add) + layernorm, D=512, wave per row ---------
__global__ void add_ln(const float* __restrict__ X, const float* __restrict__ Y,
                       const float* __restrict__ g, const float* __restrict__ be,
                       float* __restrict__ out) {
  int row  = (blockIdx.x * blockDim.x + threadIdx.x) >> 5;
  int lane = threadIdx.x & 31;
  if (row >= MROWS) return;
  const float* x = X + (size_t)row * DM;
  const float* y = Y ? (Y + (size_t)row * DM) : nullptr;
  float vals[16];
  float s = 0.f;
#pragma unroll
  for (int j = 0; j < 16; j++) {
    float v = x[lane + j * 32];
    if (y) v += y[lane + j * 32];
    vals[j] = v;
    s += v;
  }
#pragma unroll
  for (int off = 16; off; off >>= 1) s += __shfl_xor(s, off);
  float mean = s * (1.f / (float)DM);
  float var = 0.f;
#pragma unroll
  for (int j = 0; j < 16; j++) { float d = vals[j] - mean; var += d * d; }
#pragma unroll
  for (int off = 16; off; off >>= 1) var += __shfl_xor(var, off);
  var *= (1.f / (float)DM);
  float rs = rsqrtf(var + 1e-5f);
#pragma unroll
  for (int j = 0; j < 16; j++)
    out[(size_t)row * DM + lane + j * 32] =
        (vals[j] - mean) * rs * g[lane + j * 32] + be[lane + j * 32];
}

// ---------------- host-side launchers -------------------------------------
static inline void launch_gemm(const float* A, const float* Bw, const float* bias,
                               float* C, int M, int N, int K, hipStream_t s) {
  int tiles = (M / 16) * (N / 16);
  int blocks = (tiles * 32 + 255) / 256;
  gemm_wmma<<<blocks, 256, 0, s>>>(A, Bw, bias, C, M, N, K);
}
// N/16 must be a multiple of 8, K <= 512
static inline void launch_gemm_lds(const float* A, const float* Bw, const float* bias,
                                   float* C, int M, int N, int K, hipStream_t s) {
  int tiles = (M / 16) * (N / 16);
  gemm_wmma_lds<<<tiles / 8, 256, 0, s>>>(A, Bw, bias, C, M, N, K);
}

extern "C" void kernel_launch(void* const* d_in, const int* in_sizes, int n_in,
                              void* d_out, int out_size, void* d_ws, size_t ws_size,
                              hipStream_t stream) {
  (void)in_sizes; (void)n_in; (void)out_size; (void)ws_size;
  const float* src    = (const float*)d_in[0];
  const float* tok_w  = (const float*)d_in[1];
  const float* tok_b  = (const float*)d_in[2];
  const float* q_w    = (const float*)d_in[3];
  const float* q_b    = (const float*)d_in[4];
  const float* k_w    = (const float*)d_in[5];
  const float* k_b    = (const float*)d_in[6];
  const float* v_w    = (const float*)d_in[7];
  const float* v_b    = (const float*)d_in[8];
  const float* out_w  = (const float*)d_in[9];
  const float* out_b  = (const float*)d_in[10];
  const float* emb_w  = (const float*)d_in[11];
  const float* lin1_w = (const float*)d_in[12];
  const float* lin1_b = (const float*)d_in[13];
  const float* lin2_w = (const float*)d_in[14];
  const float* lin2_b = (const float*)d_in[15];
  const float* ln1_g  = (const float*)d_in[16];
  const float* ln1_b  = (const float*)d_in[17];
  const float* ln2_g  = (const float*)d_in[18];
  const float* ln2_b  = (const float*)d_in[19];
  const float* enc_g  = (const float*)d_in[20];
  const float* enc_b  = (const float*)d_in[21];
  const float* dec_w  = (const float*)d_in[22];
  const float* dec_b  = (const float*)d_in[23];
  const float* tcn_w1 = (const float*)d_in[24];
  const float* tcn_b1 = (const float*)d_in[25];
  const float* tcn_w2 = (const float*)d_in[26];
  const float* tcn_b2 = (const float*)d_in[27];

  float* w = (float*)d_ws;
  size_t off = 0;
  auto alloc = [&](size_t n) { float* p = w + off; off += n; return p; };
  float* x     = alloc((size_t)MROWS * DM);
  float* q_lin = alloc((size_t)MROWS * KD);
  float* k_lin = alloc((size_t)MROWS * KD);
  float* v_lin = alloc((size_t)MROWS * KD);
  float* qT    = alloc((size_t)64 * HD_ * L_);
  float* kT    = alloc((size_t)64 * HD_ * L_);
  float* kwb   = alloc((size_t)64 * HD_ * L_);
  float* tiF   = alloc(256);  int*  ti = (int*)tiF;
  float* tw    = alloc(256);
  float* lc    = alloc((size_t)64 * L_);
  float* tmp6  = alloc(16);
  float* fldF  = alloc(16);   int* folded = (int*)fldF;
  float* vt    = alloc((size_t)64 * 32 * L_);
  float* hbuf  = alloc((size_t)64 * 32 * L_);
  float* agg   = alloc((size_t)64 * L_ * 32);
  float* aggR  = alloc((size_t)MROWS * 256);
  float* attn  = alloc((size_t)MROWS * DM);
  float* s1    = alloc((size_t)MROWS * DM);
  float* ffn1  = alloc((size_t)MROWS * FF_);
  float* ffn2  = alloc((size_t)MROWS * DM);
  float* s2    = alloc((size_t)MROWS * DM);
  float* s3    = alloc((size_t)MROWS * DM);

  // 1. embed + positional encoding
  embed_k<<<(MROWS * DM) / 256, 256, 0, stream>>>(src, tok_w, tok_b, x);
  // 2. Q/K/V projections (WMMA, A strip staged in LDS via async copies)
  launch_gemm_lds(x, q_w, q_b, q_lin, MROWS, KD, DM, stream);
  launch_gemm_lds(x, k_w, k_b, k_lin, MROWS, KD, DM, stream);
  launch_gemm_lds(x, v_w, v_b, v_lin, MROWS, KD, DM, stream);
  // 3. repack to (bh, hd, L)
  repack_qk<<<(64 * HD_ * L_) / 256, 256, 0, stream>>>(q_lin, qT);
  repack_qk<<<(64 * HD_ * L_) / 256, 256, 0, stream>>>(k_lin, kT);
  // 4. cos-sim top-12 weights, folded k channels
  topcos_k<<<1, 16, 0, stream>>>(emb_w, ti, tw);
  build_kw<<<(64 * HD_ * L_) / 256, 256, 0, stream>>>(kT, ti, tw, kwb);
  // 5. lag correlation -> len_corr; mean + top-6 + softmax + folded sizes
  len_corr_k<<<(64 * L_ * 32) / 256, 256, 0, stream>>>(qT, kwb, lc);
  mean_topk<<<1, L_, 0, stream>>>(lc, tmp6, folded);
  // 6. v concat emb, transposed layout; zero agg
  build_v<<<(64 * L_ * 32) / 256, 256, 0, stream>>>(v_lin, emb_w, vt);
  zero_k<<<(64 * L_ * 32) / 256, 256, 0, stream>>>(agg, 64 * L_ * 32);
  // 7. six TCN temporal blocks (data-dependent kernel size, WMMA implicit GEMM)
  for (int i = 0; i < TOPK_; i++) {
    tcn_conv<<<512, 256, 0, stream>>>(vt, tcn_w1, tcn_b1, vt, hbuf, folded, tmp6, i, 0);
    tcn_conv<<<512, 256, 0, stream>>>(hbuf, tcn_w2, tcn_b2, vt, agg, folded, tmp6, i, 1);
  }
  // 8. output projection + residual + LN
  repack_agg<<<(MROWS * 256) / 256, 256, 0, stream>>>(agg, aggR);
  launch_gemm_lds(aggR, out_w, out_b, attn, MROWS, DM, 256, stream);
  add_ln<<<MROWS * 32 / 256, 256, 0, stream>>>(x, attn, ln1_g, ln1_b, s1);
  // 9. FFN (no activation in reference) + LN + encoder-final LN
  launch_gemm(s1, lin1_w, lin1_b, ffn1, MROWS, FF_, DM, stream);
  launch_gemm_lds(ffn1, lin2_w, lin2_b, ffn2, MROWS, DM, FF_, stream);
  add_ln<<<MROWS * 32 / 256, 256, 0, stream>>>(s1, ffn2, ln2_g, ln2_b, s2);
  add_ln<<<MROWS * 32 / 256, 256, 0, stream>>>(s2, nullptr, enc_g, enc_b, s3);
  // 10. decoder head -> d_out (L,B,CIN)
  launch_gemm(s3, dec_w, dec_b, (float*)d_out, MROWS, CIN, DM, stream);
}